// WeightedNTXentLoss_22084721836282
// MI455X (gfx1250) — compile-verified
//
#include <hip/hip_runtime.h>
#include <hip/hip_bf16.h>
#include <math.h>

typedef float v2f __attribute__((ext_vector_type(2)));
typedef float v8f __attribute__((ext_vector_type(8)));

#define BB   256
#define NX   256
#define TT   100
#define N2   512           // 2B
#define LAM  0.9f

// ---------------------------------------------------------------------------
// Kernel 1: build A (=a_i) and Bm (=b_j) rows [512][256] + row squared norms.
// ---------------------------------------------------------------------------
__global__ void prep_kernel(const float* __restrict__ x1,
                            const float* __restrict__ x2,
                            const float* __restrict__ u,
                            const float* __restrict__ dt,
                            const float* __restrict__ dx,
                            const float* __restrict__ vs,
                            float* __restrict__ A, float* __restrict__ Bm,
                            float* __restrict__ na, float* __restrict__ nb)
{
    const int i = blockIdx.x;          // 0..511  (row of reps = concat(x2, x1))
    const int k = threadIdx.x;         // 0..255  (spatial index)
    const int b = i & (BB - 1);        // i % B

    const float* xr = (i < BB) ? (x2 + (size_t)i * NX) : (x1 + (size_t)(i - BB) * NX);
    const float xk  = xr[k];
    const float xm1 = xr[(k + NX - 1) & (NX - 1)];   // roll(+1): u[k-1]
    const float xp1 = xr[(k + 1) & (NX - 1)];        // roll(-1): u[k+1]

    const float dxv = dx[0];
    const float dtv = dt[b];
    const float c0 = vs[b * 3 + 0];
    const float c1 = vs[b * 3 + 1];
    const float c2 = vs[b * 3 + 2];

    const float adv  = c0 * dtv * (xm1 - xp1) * xk / (2.0f * dxv);
    const float diff = -(c1 * dtv * (xm1 + xp1 - 2.0f * xk) / (dxv * dxv));
    const float dis  = -c2 * dtv * (xk - xp1) / dxv;
    const float upd  = adv + diff + dis;

    const float* urow = u + ((size_t)b * NX + k) * TT;
    const float um2 = urow[TT - 2];
    const float um1 = urow[TT - 1];

    const float av = -(um2 + xk + upd);   // a_i[k]
    const float bv = um1 + xk;            // b_i[k]

    A [(size_t)i * NX + k] = av;
    Bm[(size_t)i * NX + k] = bv;

    // block reduction of squared norms (wave32 shuffle + LDS)
    float sa = av * av, sb = bv * bv;
    #pragma unroll
    for (int off = 16; off > 0; off >>= 1) {
        sa += __shfl_down(sa, off, 32);
        sb += __shfl_down(sb, off, 32);
    }
    __shared__ float lA[8], lB[8];
    const int wid = threadIdx.x >> 5, lane = threadIdx.x & 31;
    if (lane == 0) { lA[wid] = sa; lB[wid] = sb; }
    __syncthreads();
    if (threadIdx.x == 0) {
        float ta = 0.0f, tb = 0.0f;
        #pragma unroll
        for (int w = 0; w < 8; ++w) { ta += lA[w]; tb += lB[w]; }
        na[i] = ta; nb[i] = tb;
    }
}

// ---------------------------------------------------------------------------
// Kernel 2: simmat = na[i] + nb[j] + 2 * (A . Bm^T)  via V_WMMA_F32_16X16X4_F32
// One wave per 16x16 tile, 8 waves/block, 128 blocks -> 1024 tiles.
// ---------------------------------------------------------------------------
__global__ void gemm_kernel(const float* __restrict__ A,
                            const float* __restrict__ Bm,
                            const float* __restrict__ na,
                            const float* __restrict__ nb,
                            float* __restrict__ S)
{
    const int wave = threadIdx.x >> 5;
    const int tile = blockIdx.x * 8 + wave;       // 0..1023
    const int tm = tile >> 5;                     // tile / 32
    const int tn = tile & 31;                     // tile % 32

    const int lane = threadIdx.x & 31;
    const int half = lane >> 4;                   // 0: K{0,1}, 1: K{2,3}
    const int lm   = lane & 15;

    const float* Arow = A  + (size_t)(tm * 16 + lm) * NX;   // row m of A
    const float* Brow = Bm + (size_t)(tn * 16 + lm) * NX;   // row n of B (column of B^T)

    v8f acc = {};
    #pragma unroll 8
    for (int k = 0; k < NX; k += 4) {
        v2f av = *(const v2f*)(Arow + k + half * 2);
        v2f bv = *(const v2f*)(Brow + k + half * 2);
        acc = __builtin_amdgcn_wmma_f32_16x16x4_f32(
                  false, av, false, bv, (short)0, acc, false, false);
    }

    // C/D layout: VGPR r holds M = r + 8*half, lane%16 = N
    const int n = tn * 16 + lm;
    const float nbv = nb[n];
    #pragma unroll
    for (int r = 0; r < 8; ++r) {
        const int m = tm * 16 + half * 8 + r;
        S[(size_t)m * N2 + n] = na[m] + nbv + 2.0f * acc[r];
    }
}

// ---------------------------------------------------------------------------
// Kernel 3: per-row logits -> logsumexp - pos.  One block per row i.
// ---------------------------------------------------------------------------
__global__ void row_kernel(const float* __restrict__ S,
                           const float* __restrict__ vs,
                           float* __restrict__ rowloss)
{
    const int i  = blockIdx.x;                    // 0..511
    const int t  = threadIdx.x;                   // 0..255
    const int ib = i & (BB - 1);
    const int jp = (i < BB) ? (i + BB) : (i - BB);  // positive partner

    const float vi0 = vs[ib * 3 + 0], vi1 = vs[ib * 3 + 1], vi2 = vs[ib * 3 + 2];
    const float nvi = sqrtf(vi0 * vi0 + vi1 * vi1 + vi2 * vi2);

    __shared__ float sm[8], ssum[8], spos;

    float vals[2];
    float lmax = -INFINITY;
    #pragma unroll
    for (int q = 0; q < 2; ++q) {
        const int j = t + q * 256;
        float v;
        if (j == i) {
            v = -INFINITY;                        // excluded (self)
        } else {
            const float sv = S[(size_t)i * N2 + j];
            if (j == jp) {
                v = sv;                           // positive logit (weight 1)
                spos = sv;                        // exactly one writer per row
            } else {
                const int jb = j & (BB - 1);
                const float vj0 = vs[jb * 3 + 0], vj1 = vs[jb * 3 + 1], vj2 = vs[jb * 3 + 2];
                const float nvj  = sqrtf(vj0 * vj0 + vj1 * vj1 + vj2 * vj2);
                const float prod = sqrtf(fabsf(vi0 * vj0) + fabsf(vi1 * vj1) + fabsf(vi2 * vj2));
                const float sim  = prod / fmaxf(nvi, nvj);
                float w = 1.0f - LAM * sim;
                if (isnan(w)) w = 0.0f;           // nan_to_num(..., nan=0)
                v = sv * w;
            }
        }
        vals[q] = v;
        lmax = fmaxf(lmax, v);
    }

    // block max
    #pragma unroll
    for (int off = 16; off > 0; off >>= 1) lmax = fmaxf(lmax, __shfl_down(lmax, off, 32));
    const int wid = t >> 5, lane = t & 31;
    if (lane == 0) sm[wid] = lmax;
    __syncthreads();
    float gmax = sm[0];
    #pragma unroll
    for (int w = 1; w < 8; ++w) gmax = fmaxf(gmax, sm[w]);

    // block sum of exp(v - gmax); expf(-inf) = 0 handles exclusions
    float lsum = expf(vals[0] - gmax) + expf(vals[1] - gmax);
    #pragma unroll
    for (int off = 16; off > 0; off >>= 1) lsum += __shfl_down(lsum, off, 32);
    if (lane == 0) ssum[wid] = lsum;
    __syncthreads();
    if (t == 0) {
        float tot = 0.0f;
        #pragma unroll
        for (int w = 0; w < 8; ++w) tot += ssum[w];
        rowloss[i] = gmax + logf(tot) - spos;     // lse_i - pos_i
    }
}

// ---------------------------------------------------------------------------
// Kernel 4: deterministic final reduction: out = sum(rowloss) / 512
// ---------------------------------------------------------------------------
__global__ void final_kernel(const float* __restrict__ rowloss, float* __restrict__ out)
{
    const int t = threadIdx.x;                    // 256 threads
    float s = rowloss[t] + rowloss[t + 256];
    #pragma unroll
    for (int off = 16; off > 0; off >>= 1) s += __shfl_down(s, off, 32);
    __shared__ float sw[8];
    const int wid = t >> 5, lane = t & 31;
    if (lane == 0) sw[wid] = s;
    __syncthreads();
    if (t == 0) {
        float tot = 0.0f;
        #pragma unroll
        for (int w = 0; w < 8; ++w) tot += sw[w];
        out[0] = tot / (float)N2;
    }
}

// ---------------------------------------------------------------------------
extern "C" void kernel_launch(void* const* d_in, const int* in_sizes, int n_in,
                              void* d_out, int out_size, void* d_ws, size_t ws_size,
                              hipStream_t stream)
{
    const float* x1 = (const float*)d_in[0];
    const float* x2 = (const float*)d_in[1];
    const float* u  = (const float*)d_in[2];
    const float* dt = (const float*)d_in[3];
    const float* dx = (const float*)d_in[4];
    const float* vs = (const float*)d_in[5];
    float* out = (float*)d_out;

    // workspace layout (floats): A[512*256] Bm[512*256] na[512] nb[512] S[512*512] rl[512]
    float* A   = (float*)d_ws;
    float* Bm  = A  + (size_t)N2 * NX;
    float* na  = Bm + (size_t)N2 * NX;
    float* nb  = na + N2;
    float* S   = nb + N2;
    float* rl  = S  + (size_t)N2 * N2;

    prep_kernel <<<N2, 256, 0, stream>>>(x1, x2, u, dt, dx, vs, A, Bm, na, nb);
    gemm_kernel <<<128, 256, 0, stream>>>(A, Bm, na, nb, S);
    row_kernel  <<<N2, 256, 0, stream>>>(S, vs, rl);
    final_kernel<<<1, 256, 0, stream>>>(rl, out);
}